// QuantizedGRU_5196910428519
// MI455X (gfx1250) — compile-verified
//
#include <hip/hip_runtime.h>

// Quantized GRU (T=512, B=1024, H=12), Q8.7 fixed point.
// One wave32 per 16 batch rows. Entire input slab for the tile is preloaded
// into LDS via GLOBAL_LOAD_ASYNC_TO_LDS_B128 (ASYNCcnt) so the 512-step
// serial recurrence never touches global memory except the output store.
// h@W_hh.T via V_WMMA_F32_16X16X4_F32 (3 gate N-tiles x 3 K=4 chunks).

#define HH 12          // hidden size
#define GH 16          // padded hidden row stride in LDS
#define TILE_M 16      // batch rows per wave
#define TMAX 512       // max timesteps supported by LDS slab
#define QSCALE 128.0f
#define QINV   0.0078125f

typedef float v2f __attribute__((ext_vector_type(2)));
typedef float v8f __attribute__((ext_vector_type(8)));
typedef int   v4i __attribute__((ext_vector_type(4)));

#define AS1 __attribute__((address_space(1)))
#define AS3 __attribute__((address_space(3)))

#if defined(__gfx1250__) && __has_builtin(__builtin_amdgcn_global_load_async_to_lds_b128)
#define ASYNC_LDS 1
#else
#define ASYNC_LDS 0
#endif

__device__ __forceinline__ float qz(float x) {
  // clip(round(x*128), -2^15, 2^15-1)/128 ; rintf == RNE == jnp.round
  float r = rintf(x * QSCALE);
  r = fminf(fmaxf(r, -32768.0f), 32767.0f);
  return r * QINV;
}

__device__ __forceinline__ float hsig(float x) {
  return fminf(fmaxf(0.25f * x + 0.5f, 0.0f), 1.0f);
}

__global__ void __launch_bounds__(32)
qgru_wmma_kernel(const float* __restrict__ in,     // [T,B,1]
                 const float* __restrict__ w_ih,   // [3H,1]
                 const float* __restrict__ w_hh,   // [3H,H]
                 const float* __restrict__ b_ih,   // [3H]
                 const float* __restrict__ b_hh,   // [3H]
                 const float* __restrict__ w1,     // [H/2,H]
                 const float* __restrict__ b1,     // [H/2]
                 const float* __restrict__ w2,     // [1,H/2]
                 const float* __restrict__ b2,     // [1]
                 float* __restrict__ out,          // [T,B,1]
                 int T, int B)
{
  __shared__ float xslab[TMAX * TILE_M];  // x[t][row], row-contiguous (32 KB)
  __shared__ float h_lds[TILE_M][GH];     // hidden state tile (cols 12..15 unused)

  const int lane = threadIdx.x;           // 0..31
  const int m    = lane & 15;
  const int hi   = lane >> 4;
  const int base = blockIdx.x * TILE_M;

  // ---- init h = 0 ----
  if (lane < TILE_M) {
#pragma unroll
    for (int k = 0; k < GH; ++k) h_lds[lane][k] = 0.0f;
  }

  // ---- preload input slab: rows [base, base+16) for all T timesteps ----
  // chunk c covers 8 timesteps; lane l copies 16B: t = c*8 + l/4, col = (l%4)*4
  {
    const int nchunk = T / 8;             // 64 for T=512
    const int tloc = lane >> 2;
    const int rloc = (lane & 3) * 4;
#if ASYNC_LDS
    for (int c = 0; c < nchunk; ++c) {
      const int tt = c * 8 + tloc;
      const float* g = in + (size_t)tt * B + base + rloc;
      float* l = &xslab[tt * TILE_M + rloc];
      __builtin_amdgcn_global_load_async_to_lds_b128(
          (AS1 v4i*)g, (AS3 v4i*)l, 0, 0);
    }
#if __has_builtin(__builtin_amdgcn_s_wait_asynccnt)
    __builtin_amdgcn_s_wait_asynccnt(0);
#else
    asm volatile("s_wait_asynccnt 0x0" ::: "memory");
#endif
#else
    for (int c = 0; c < nchunk; ++c) {
      const int tt = c * 8 + tloc;
      float4 v = *(const float4*)(in + (size_t)tt * B + base + rloc);
      *(float4*)&xslab[tt * TILE_M + rloc] = v;
    }
#endif
  }

  // ---- per-lane gate-column constants (col n = m within each gate) ----
  float wihq[3], bihq[3], bhhq[3];
#pragma unroll
  for (int g = 0; g < 3; ++g) {
    if (m < HH) {
      wihq[g] = qz(w_ih[g * HH + m]);
      bihq[g] = qz(b_ih[g * HH + m]);
      bhhq[g] = qz(b_hh[g * HH + m]);
    } else {
      wihq[g] = 0.0f; bihq[g] = 0.0f; bhhq[g] = 0.0f;
    }
  }

  // ---- W_hh^T in WMMA B-layout: 3 gate tiles x 3 K-chunks, in registers ----
  // B 4x16 f32: VGPR0 lanes0-15 = K=4c, lanes16-31 = K=4c+2; VGPR1 = +1
  v2f bB[3][3];
#pragma unroll
  for (int g = 0; g < 3; ++g) {
#pragma unroll
    for (int c = 0; c < 3; ++c) {
      int k0 = 4 * c + 2 * hi;            // k0, k0+1 < 12 for c < 3
      float bx = 0.0f, by = 0.0f;
      if (m < HH) {
        bx = qz(w_hh[(g * HH + m) * HH + k0]);
        by = qz(w_hh[(g * HH + m) * HH + k0 + 1]);
      }
      bB[g][c].x = bx;
      bB[g][c].y = by;
    }
  }

  // ---- FCNN weights split across half-waves: this half owns j = hi*3 .. hi*3+2
  const int j0 = hi * 3;
  float w1q[3][HH], b1q[3], w2q[3], b2q;
#pragma unroll
  for (int jj = 0; jj < 3; ++jj) {
#pragma unroll
    for (int k = 0; k < HH; ++k) w1q[jj][k] = qz(w1[(j0 + jj) * HH + k]);
    b1q[jj] = qz(b1[j0 + jj]);
    w2q[jj] = qz(w2[j0 + jj]);
  }
  b2q = qz(b2[0]);

  __syncthreads();

  // h_old in registers: C-layout element (row=i+8*hi, col=m) owned by this lane
  float hreg[8];
#pragma unroll
  for (int i = 0; i < 8; ++i) hreg[i] = 0.0f;

  for (int t = 0; t < T; ++t) {
    // ---- A fragments of h from LDS (cross-lane): lane row M=m, K=4c+2*hi ----
    v2f aA[3];
#pragma unroll
    for (int c = 0; c < 3; ++c) {
      int k0 = 4 * c + 2 * hi;
      aA[c].x = h_lds[m][k0];
      aA[c].y = h_lds[m][k0 + 1];
    }

    // x for this half's 8 rows, from the LDS slab (2x ds_load_b128)
    float xv8[8];
    *(float4*)&xv8[0] = *(const float4*)&xslab[t * TILE_M + 8 * hi];
    *(float4*)&xv8[4] = *(const float4*)&xslab[t * TILE_M + 8 * hi + 4];

    // ---- hh = h @ W_hh^T per gate via chained WMMA (K = 12) ----
    v8f acc[3];
#pragma unroll
    for (int g = 0; g < 3; ++g) {
      v8f c = {};
#pragma unroll
      for (int kc = 0; kc < 3; ++kc) {
        c = __builtin_amdgcn_wmma_f32_16x16x4_f32(
            /*neg_a=*/false, aA[kc], /*neg_b=*/false, bB[g][kc],
            /*c_mod=*/(short)0, c, /*reuse_a=*/false, /*reuse_b=*/false);
      }
      acc[g] = c;
    }

    // ---- gates + state update in C-layout: VGPR i -> (row=i+8*hi, col=m) ----
#pragma unroll
    for (int i = 0; i < 8; ++i) {
      int row = i + 8 * hi;
      float xv  = xv8[i];
      float hhr = qz(qz(acc[0][i]) + bhhq[0]);
      float hhz = qz(qz(acc[1][i]) + bhhq[1]);
      float hhn = qz(qz(acc[2][i]) + bhhq[2]);
      float iir = qz(qz(xv * wihq[0]) + bihq[0]);
      float iiz = qz(qz(xv * wihq[1]) + bihq[1]);
      float iin = qz(qz(xv * wihq[2]) + bihq[2]);
      float r  = qz(hsig(iir + hhr));
      float z  = qz(hsig(iiz + hhz));
      float nt = qz(fminf(fmaxf(iin + r * hhn, -1.0f), 1.0f));
      float hn = qz((1.0f - z) * nt + z * hreg[i]);  // == 0 in padding cols
      hreg[i] = hn;
      h_lds[row][m] = hn;
    }
    __syncthreads();

    // ---- FCNN on h_new: 12 -> 6 -> 1; each half-wave owns 3 of the 6 mids ----
    {
      float4 ha = *(const float4*)&h_lds[m][0];
      float4 hb = *(const float4*)&h_lds[m][4];
      float4 hc = *(const float4*)&h_lds[m][8];
      float hrow[HH] = { ha.x, ha.y, ha.z, ha.w,
                         hb.x, hb.y, hb.z, hb.w,
                         hc.x, hc.y, hc.z, hc.w };
      float part = 0.0f;
#pragma unroll
      for (int jj = 0; jj < 3; ++jj) {
        float s = 0.0f;
#pragma unroll
        for (int k = 0; k < HH; ++k) s += hrow[k] * w1q[jj][k];
        s = qz(s);
        s = qz(s + b1q[jj]);                    // extra qz() is a no-op
        part += fmaxf(s, 0.0f) * w2q[jj];
      }
      float tot = part + __shfl_xor(part, 16, 32);
      if (lane < TILE_M) {
        float o = qz(tot);
        o = qz(o + b2q);
        out[(size_t)t * B + base + m] = o;
      }
    }
    __syncthreads();
  }
}

extern "C" void kernel_launch(void* const* d_in, const int* in_sizes, int n_in,
                              void* d_out, int out_size, void* d_ws, size_t ws_size,
                              hipStream_t stream) {
  (void)n_in; (void)d_ws; (void)ws_size; (void)out_size;
  const float* in   = (const float*)d_in[0];
  const float* w_ih = (const float*)d_in[1];
  const float* w_hh = (const float*)d_in[2];
  const float* b_ih = (const float*)d_in[3];
  const float* b_hh = (const float*)d_in[4];
  const float* w1   = (const float*)d_in[5];
  const float* b1   = (const float*)d_in[6];
  const float* w2   = (const float*)d_in[7];
  const float* b2   = (const float*)d_in[8];
  float* out = (float*)d_out;

  const int B = 1024;                 // fixed by the reference harness
  int T = in_sizes[0] / B;            // input is [T,B,1]
  if (T > TMAX) T = TMAX;

  dim3 grid(B / TILE_M);              // 64 waves, one 16-row tile each
  dim3 block(32);                     // one wave32 per block
  qgru_wmma_kernel<<<grid, block, 0, stream>>>(in, w_ih, w_hh, b_ih, b_hh,
                                               w1, b1, w2, b2, out, T, B);
}